// Net_27882927686344
// MI455X (gfx1250) — compile-verified
//
#include <hip/hip_runtime.h>
#include <math.h>

// Problem constants (from reference)
#define NODES 65536
#define BGR   128
#define NPG   512
#define EDGES 524288
#define F_IN  128
#define H1    64
#define H2    128
#define NCLS  10
#define KTOP  410   // ceil(0.8*512)

typedef float v2f __attribute__((ext_vector_type(2)));
typedef float v8f __attribute__((ext_vector_type(8)));

// ---------------------------------------------------------------- init / degree
__global__ void k_set_ones(float* p, int n) {
  int i = blockIdx.x * blockDim.x + threadIdx.x;
  if (i < n) p[i] = 1.0f;                    // self-loop contributes 1 to degree
}

__global__ void k_degree(const int* __restrict__ dst, float* __restrict__ deg, int E) {
  int e = blockIdx.x * blockDim.x + threadIdx.x;
  if (e < E) atomicAdd(&deg[dst[e]], 1.0f);
}

__global__ void k_rsqrt_inplace(float* p, int n) {
  int i = blockIdx.x * blockDim.x + threadIdx.x;
  if (i < n) p[i] = rsqrtf(p[i]);            // deg >= 1 always (self loop)
}

// ---------------------------------------------------------------- WMMA GEMM
// C[M x NDIM] = A[M x KDIM] @ B[KDIM x NDIM], all f32 row-major.
// One wave32 computes a 16-row strip across all NDIM columns using
// V_WMMA_F32_16X16X4_F32 (fp32 A/B, fp32 accumulate — matches reference math).
// B panel staged in LDS (32 KB) with +2 float row pad to avoid bank conflicts.
template <int KDIM, int NDIM>
__global__ __launch_bounds__(256) void k_gemm_wmma(const float* __restrict__ A,
                                                   const float* __restrict__ B,
                                                   float* __restrict__ C, int M) {
  constexpr int BSTRIDE = NDIM + 2;
  __shared__ float Bs[KDIM * BSTRIDE];
  const int tid = threadIdx.x;
  for (int i = tid; i < KDIM * NDIM; i += 256) {
    int r = i / NDIM, c = i % NDIM;
    Bs[r * BSTRIDE + c] = B[i];
  }
  __syncthreads();

  const int lane = tid & 31;
  const int wave = tid >> 5;
  const int row0 = (blockIdx.x * 8 + wave) * 16;
  if (row0 >= M) return;

  const int m  = lane & 15;    // row within A frag / col within B,C frags
  const int kh = lane >> 4;    // which K-half this lane holds

  constexpr int NT = NDIM / 16;
  v8f acc[NT];
#pragma unroll
  for (int t = 0; t < NT; ++t) acc[t] = (v8f){0.f, 0.f, 0.f, 0.f, 0.f, 0.f, 0.f, 0.f};

  const float* Arow = A + (size_t)(row0 + m) * KDIM;
#pragma unroll 4
  for (int k0 = 0; k0 < KDIM; k0 += 4) {
    const int ka = k0 + 2 * kh;                 // even -> 8B aligned
    const float2 av = *(const float2*)(Arow + ka);
    v2f a; a.x = av.x; a.y = av.y;
#pragma unroll
    for (int t = 0; t < NT; ++t) {
      const int c = t * 16 + m;
      v2f b;
      b.x = Bs[ka * BSTRIDE + c];
      b.y = Bs[(ka + 1) * BSTRIDE + c];
      // (neg_a, A, neg_b, B, c_mod, C, reuse_a, reuse_b)
      acc[t] = __builtin_amdgcn_wmma_f32_16x16x4_f32(
          false, a, false, b, (short)0, acc[t], false, false);
    }
  }

#pragma unroll
  for (int t = 0; t < NT; ++t)
#pragma unroll
    for (int i = 0; i < 8; ++i)
      C[(size_t)(row0 + i + 8 * kh) * NDIM + t * 16 + m] = acc[t][i];
}

// ---------------------------------------------------------------- edge scatter
// agg[dst] += dis[src]*dis[dst] * h[src]  (float4 gathers, f32 L2 atomics)
template <int H>
__global__ void k_scatter(const float* __restrict__ h, const int* __restrict__ src,
                          const int* __restrict__ dst, const float* __restrict__ dis,
                          float* __restrict__ agg, int E) {
  int e = blockIdx.x * blockDim.x + threadIdx.x;
  if (e >= E) return;
  const int s = src[e], d = dst[e];
  const float nm = dis[s] * dis[d];
  const float4* hs = (const float4*)(h + (size_t)s * H);
  float* ad = agg + (size_t)d * H;
#pragma unroll
  for (int q = 0; q < H / 4; ++q) {
    float4 v = hs[q];
    atomicAdd(&ad[4 * q + 0], nm * v.x);
    atomicAdd(&ad[4 * q + 1], nm * v.y);
    atomicAdd(&ad[4 * q + 2], nm * v.z);
    atomicAdd(&ad[4 * q + 3], nm * v.w);
  }
}

// self-loop term + bias + ReLU, in place on agg
template <int H>
__global__ void k_finalize(const float* __restrict__ h, const float* __restrict__ dis,
                           const float* __restrict__ bias, float* __restrict__ agg,
                           int total) {
  int i = blockIdx.x * blockDim.x + threadIdx.x;
  if (i >= total) return;
  const int node = i / H, f = i % H;
  const float d = dis[node];
  const float v = agg[i] + d * d * h[i] + bias[f];
  agg[i] = v > 0.f ? v : 0.f;
}

// ---------------------------------------------------------------- TopK pool
// One block per graph: score -> bitonic sort (threshold = 410th largest) ->
// tanh gate -> per-feature gated max over selected nodes.
__global__ __launch_bounds__(512) void k_pool(const float* __restrict__ x2,
                                              const float* __restrict__ pw,
                                              float* __restrict__ g) {
  __shared__ float sc[NPG];
  __shared__ float srt[NPG];
  __shared__ float gate[NPG];
  const int t = threadIdx.x;
  const int b = blockIdx.x;

  float wn = 0.f;
  for (int f = 0; f < H2; ++f) { float w = pw[f]; wn += w * w; }
  wn = sqrtf(wn);

  const float* xr = x2 + ((size_t)b * NPG + t) * H2;
  float s = 0.f;
  for (int f = 0; f < H2; ++f) s += xr[f] * pw[f];
  s /= wn;
  sc[t] = s;
  srt[t] = s;
  __syncthreads();

  // ascending bitonic sort of 512 scores
  for (int k = 2; k <= NPG; k <<= 1) {
    for (int j = k >> 1; j > 0; j >>= 1) {
      int ixj = t ^ j;
      if (ixj > t) {
        float a0 = srt[t], a1 = srt[ixj];
        bool up = ((t & k) == 0);
        if ((a0 > a1) == up) { srt[t] = a1; srt[ixj] = a0; }
      }
      __syncthreads();
    }
  }
  const float thresh = srt[NPG - KTOP];   // 410th largest
  gate[t] = tanhf(sc[t]);
  __syncthreads();

  // gated max: 4 node-chunks x 128 features
  const int f = t & (H2 - 1);
  const int chunk = t >> 7;
  float m = -INFINITY;
  const float* xb = x2 + (size_t)b * NPG * H2;
  for (int i = 0; i < NPG / 4; ++i) {
    int n = chunk * (NPG / 4) + i;
    if (sc[n] >= thresh) m = fmaxf(m, xb[(size_t)n * H2 + f] * gate[n]);
  }
  srt[t] = m;   // reuse as partial-max buffer
  __syncthreads();
  if (t < H2) {
    float r = fmaxf(fmaxf(srt[t], srt[t + 128]), fmaxf(srt[t + 256], srt[t + 384]));
    g[(size_t)b * H2 + t] = r;
  }
}

// ---------------------------------------------------------------- FC + log_softmax
__global__ __launch_bounds__(32) void k_classify(const float* __restrict__ g,
                                                 const float* __restrict__ fcW,
                                                 const float* __restrict__ fcb,
                                                 float* __restrict__ out) {
  const int b = blockIdx.x, lane = threadIdx.x;
  __shared__ float lg[NCLS];
  if (lane < NCLS) {
    float s = fcb[lane];
    const float* gr = g + (size_t)b * H2;
    for (int f = 0; f < H2; ++f) s += gr[f] * fcW[f * NCLS + lane];
    lg[lane] = s;
  }
  __syncthreads();
  if (lane < NCLS) {
    float mx = lg[0];
    for (int i = 1; i < NCLS; ++i) mx = fmaxf(mx, lg[i]);
    float se = 0.f;
    for (int i = 0; i < NCLS; ++i) se += expf(lg[i] - mx);
    out[b * NCLS + lane] = lg[lane] - mx - logf(se);
  }
}

// ---------------------------------------------------------------- launch
extern "C" void kernel_launch(void* const* d_in, const int* in_sizes, int n_in,
                              void* d_out, int out_size, void* d_ws, size_t ws_size,
                              hipStream_t stream) {
  (void)in_sizes; (void)n_in; (void)out_size; (void)ws_size;
  const float* x   = (const float*)d_in[0];
  const int*   ei  = (const int*)d_in[1];   // edge_index [2, E] (harness int convention)
  // d_in[2] = batch: implicit (N nodes per graph), unused
  const float* W1  = (const float*)d_in[3];
  const float* b1  = (const float*)d_in[4];
  const float* W2  = (const float*)d_in[5];
  const float* b2  = (const float*)d_in[6];
  const float* pw  = (const float*)d_in[7];
  const float* fcW = (const float*)d_in[8];
  const float* fcb = (const float*)d_in[9];
  float* out = (float*)d_out;

  float* ws  = (float*)d_ws;
  float* dis = ws;                              // NODES          (deg -> rsqrt in place)
  float* h1  = dis + NODES;                     // NODES*H1
  float* a1  = h1 + (size_t)NODES * H1;         // NODES*H1       (relu1 after finalize)
  float* h2  = a1 + (size_t)NODES * H1;         // NODES*H2
  float* a2  = h2 + (size_t)NODES * H2;         // NODES*H2       (relu2 = pooled input)
  float* g   = a2 + (size_t)NODES * H2;         // BGR*H2

  const int* src = ei;
  const int* dst = ei + EDGES;

  hipMemsetAsync(a1, 0, sizeof(float) * (size_t)NODES * H1, stream);
  hipMemsetAsync(a2, 0, sizeof(float) * (size_t)NODES * H2, stream);

  k_set_ones<<<NODES / 256, 256, 0, stream>>>(dis, NODES);
  k_degree<<<EDGES / 256, 256, 0, stream>>>(dst, dis, EDGES);
  k_rsqrt_inplace<<<NODES / 256, 256, 0, stream>>>(dis, NODES);

  // layer 1: h1 = x @ W1 ; aggregate ; relu(+b1)
  k_gemm_wmma<F_IN, H1><<<NODES / 128, 256, 0, stream>>>(x, W1, h1, NODES);
  k_scatter<H1><<<EDGES / 256, 256, 0, stream>>>(h1, src, dst, dis, a1, EDGES);
  k_finalize<H1><<<(NODES * H1) / 256, 256, 0, stream>>>(h1, dis, b1, a1, NODES * H1);

  // layer 2: h2 = relu1 @ W2 ; aggregate ; relu(+b2)
  k_gemm_wmma<H1, H2><<<NODES / 128, 256, 0, stream>>>(a1, W2, h2, NODES);
  k_scatter<H2><<<EDGES / 256, 256, 0, stream>>>(h2, src, dst, dis, a2, EDGES);
  k_finalize<H2><<<(NODES * H2) / 256, 256, 0, stream>>>(h2, dis, b2, a2, NODES * H2);

  k_pool<<<BGR, NPG, 0, stream>>>(a2, pw, g);
  k_classify<<<BGR, 32, 0, stream>>>(g, fcW, fcb, out);
}